// SingleModalFactorization_55834574848304
// MI455X (gfx1250) — compile-verified
//
#include <hip/hip_runtime.h>

// Problem constants (from reference): S=2048 samples, C=8 components,
// CH=4096 channels, all float32. inputs (d_in[0]) is UNUSED by the reference.
#define S_TOTAL 2048
#define C_COMP  8
#define CH      4096
#define G       32                 // zero-guard floats on each side of a row
#define ROWS    (CH + 2 * G)       // padded LDS row stride in floats (4160)
#define SPB     8                  // samples per block
#define TPB     512                // threads per block (16 wave32s)
#define CHPT    (CH / TPB)         // 8 channels per thread

typedef float v4f __attribute__((ext_vector_type(4)));

static __device__ __forceinline__ int clampi(int x, int lo, int hi) {
    return x < lo ? lo : (x > hi ? hi : x);
}

// Constant-index FMA body: sub-alignment R is block-uniform (scalar branch).
#define ACC_CASE(R)                                                         \
    {                                                                       \
        _Pragma("unroll")                                                   \
        for (int k = 0; k < CHPT; ++k)                                      \
            acc[k] = fmaf(A0, w12[(R) + k], fmaf(A1, w12[(R) + k + 1], acc[k])); \
    }

__global__ __launch_bounds__(TPB) void shifted_mix_kernel(
    const float* __restrict__ components,     // [C_COMP, CH]
    const float* __restrict__ contributions,  // [S_TOTAL, C_COMP]
    const float* __restrict__ shift,          // [S_TOTAL, C_COMP]
    float* __restrict__ out)                  // [S_TOTAL, CH]
{
    // Dynamic LDS: 8 padded component rows (G zeros | 4096 data | G zeros),
    // then [SPB*C_COMP] contrib + [SPB*C_COMP] shift. ~130.5 KB total.
    extern __shared__ float smem[];
    const int tid = threadIdx.x;
    const int s0  = blockIdx.x * SPB;

    // ---- Zero the guard pads (8 rows x 64 pad floats = 512 threads) ----
    {
        int row = tid >> 6;              // 0..7
        int idx = tid & 63;              // 0..63: [0,32) left pad, [32,64) right pad
        float* rowp = smem + row * ROWS;
        rowp[(idx < G) ? idx : (CH + idx)] = 0.0f;   // CH+idx in [CH+G, CH+2G)
    }

    // ---- Async-DMA the components table (128 KB) into the padded rows ----
    // CDNA5 GVS async global->LDS b128: address = SGPR base + 32-bit VGPR offset.
    {
        const uint32_t  lds_base = (uint32_t)(uintptr_t)(&smem[0]);
        const uintptr_t gbase    = (uintptr_t)components;
        #pragma unroll
        for (int it = 0; it < (C_COMP * CH * 4) / (16 * TPB); ++it) {  // 16 iters
            int ci = it * TPB + tid;          // 16B chunk id, 0..8191
            int c  = ci >> 10;                // source row (1024 chunks/row)
            int ir = ci & 1023;               // chunk within row
            uint32_t gb = (uint32_t)ci * 16u;
            uint32_t lb = lds_base + (uint32_t)(c * ROWS + G + ir * 4) * 4u;
            asm volatile("global_load_async_to_lds_b128 %0, %1, %2"
                         :: "v"(lb), "v"(gb), "s"(gbase)
                         : "memory");
        }
    }

    // Small per-block tables through the normal path.
    float* s_ct = smem + C_COMP * ROWS;          // [SPB*C_COMP] contributions
    float* s_sh = s_ct + SPB * C_COMP;           // [SPB*C_COMP] shifts
    if (tid < SPB * C_COMP) {
        s_ct[tid] = contributions[s0 * C_COMP + tid];
        s_sh[tid] = shift[s0 * C_COMP + tid];
    }

    asm volatile("s_wait_asynccnt 0" ::: "memory");
    __syncthreads();

    // ---- Per sample: 2-tap lerp gather from padded LDS + C=8 reduction ----
    const int i0 = tid * CHPT;                   // i0 % 4 == 0
    for (int sp = 0; sp < SPB; ++sp) {
        float a0[C_COMP], a1[C_COMP];
        int   lo[C_COMP];
        #pragma unroll
        for (int c = 0; c < C_COMP; ++c) {
            float sh = s_sh[sp * C_COMP + c];
            float fl = floorf(sh);
            float w  = sh - fl;                  // in [0,1)
            float ct = s_ct[sp * C_COMP + c];
            // Clamp so int math is safe; |lo|>CH+1 contributes zero either way.
            lo[c] = (int)fminf(fmaxf(fl, -(float)(CH + 2)), (float)(CH + 2));
            a0[c] = ct * (1.0f - w);
            a1[c] = ct * w;
        }

        float acc[CHPT];
        #pragma unroll
        for (int k = 0; k < CHPT; ++k) acc[k] = 0.0f;

        #pragma unroll
        for (int c = 0; c < C_COMP; ++c) {
            const float* cm = smem + c * ROWS + G;   // channel-0 pointer
            const int   l  = lo[c];
            const float A0 = a0[c], A1 = a1[c];

            if (l >= -G && l < G) {
                // FAST PATH (block-uniform condition; always taken for N(0,3) shifts):
                // 3 aligned ds_load_b128 covering taps [base, base+8]; guard pads
                // supply the zero-padding, so no masks needed.
                int base_al = i0 + (l & ~3);         // 16B-aligned float index
                int r       = l & 3;                 // block-uniform sub-alignment
                const float* cp = cm + base_al;      // in [-G, CH+G-12]
                v4f f0 = *(const v4f*)(cp + 0);
                v4f f1 = *(const v4f*)(cp + 4);
                v4f f2 = *(const v4f*)(cp + 8);
                float w12[12] = { f0.x, f0.y, f0.z, f0.w,
                                  f1.x, f1.y, f1.z, f1.w,
                                  f2.x, f2.y, f2.z, f2.w };
                switch (r) {                          // scalar (uniform) branch
                    case 0:  ACC_CASE(0); break;
                    case 1:  ACC_CASE(1); break;
                    case 2:  ACC_CASE(2); break;
                    default: ACC_CASE(3); break;
                }
            } else {
                // EXACT FALLBACK for pathological shifts: masked scalar gather.
                #pragma unroll
                for (int k = 0; k < CHPT; ++k) {
                    int j0 = i0 + k + l;
                    int j1 = j0 + 1;
                    int jc0 = clampi(j0, 0, CH - 1);
                    int jc1 = clampi(j1, 0, CH - 1);
                    float x0 = cm[jc0];
                    float x1 = cm[jc1];
                    x0 = (j0 == jc0) ? x0 : 0.0f;
                    x1 = (j1 == jc1) ? x1 : 0.0f;
                    acc[k] = fmaf(A0, x0, fmaf(A1, x1, acc[k]));
                }
            }
        }

        // Write-once output stream: non-temporal 128-bit stores.
        float* op = out + (size_t)(s0 + sp) * CH + i0;
        #pragma unroll
        for (int k = 0; k < CHPT; k += 4) {
            v4f o = { acc[k], acc[k + 1], acc[k + 2], acc[k + 3] };
            __builtin_nontemporal_store(o, (v4f*)(op + k));
        }
    }
}

extern "C" void kernel_launch(void* const* d_in, const int* in_sizes, int n_in,
                              void* d_out, int out_size, void* d_ws, size_t ws_size,
                              hipStream_t stream) {
    // setup_inputs() order: inputs(0, UNUSED), components(1), contributions(2), shift(3)
    const float* components    = (const float*)d_in[1];
    const float* contributions = (const float*)d_in[2];
    const float* shift         = (const float*)d_in[3];
    float* out = (float*)d_out;

    const size_t smem_bytes = (size_t)(C_COMP * ROWS + 2 * SPB * C_COMP) * sizeof(float);
    shifted_mix_kernel<<<dim3(S_TOTAL / SPB), dim3(TPB), smem_bytes, stream>>>(
        components, contributions, shift, out);
}